// ReservoirLayer_13838384627927
// MI455X (gfx1250) — compile-verified
//
#include <hip/hip_runtime.h>

// ---------------------------------------------------------------------------
// out = x . (0.9 I + 0.1 W)^65 . w_out^T
//   x: (16384, 512) f32,  W: (512,512) f32,  w_out: (256,512) f32
// Strategy: repeated squaring of A=0.9I+0.1W (6 squarings + 1 mul = A^65),
// fold the readout into P = A^65 . w_out^T (512x256), then one big GEMM.
// All matrix math on V_WMMA_F32_16X16X4_F32 (CDNA5 f32 matrix path).
// ---------------------------------------------------------------------------

typedef float v2f_t __attribute__((ext_vector_type(2)));
typedef float v8f_t __attribute__((ext_vector_type(8)));

#define NDIM 512
#define BDIM 16384
#define ODIM 256

static __device__ __forceinline__ v8f_t wmma_f32(v2f_t a, v2f_t b, v8f_t c) {
    // 8 args: (neg_a, A, neg_b, B, c_mod, C, reuse_a, reuse_b)
    return __builtin_amdgcn_wmma_f32_16x16x4_f32(
        false, a, false, b, (short)0, c, false, false);
}

// A = 0.9*I + 0.1*W   (512x512)
__global__ void build_A_kernel(const float* __restrict__ W, float* __restrict__ A) {
    int idx = blockIdx.x * blockDim.x + threadIdx.x;   // 512*512 threads
    int i = idx >> 9;
    int j = idx & (NDIM - 1);
    float v = 0.1f * W[idx];
    if (i == j) v += 0.9f;
    A[idx] = v;
}

// C = A * B, all 512x512 row-major f32. One 16x16 tile per wave.
// grid: 256 blocks * 4 waves = 1024 tiles.
__global__ void gemm_nn_512_kernel(const float* __restrict__ A,
                                   const float* __restrict__ B,
                                   float* __restrict__ C) {
    const int lane = threadIdx.x & 31;
    const int wave = threadIdx.x >> 5;
    const int tile = blockIdx.x * 4 + wave;       // 0..1023
    const int m0 = (tile >> 5) << 4;              // tile row * 16
    const int n0 = (tile & 31) << 4;              // tile col * 16
    const int lm = lane & 15;
    const int hi = lane >> 4;                     // 0 or 1

    v8f_t acc = {};
    const float* arow = A + (m0 + lm) * NDIM;     // row of A for this lane
    const float* bcol = B + n0 + lm;              // column base of B

#pragma unroll 4
    for (int kk = 0; kk < NDIM; kk += 4) {
        const int kb = kk + 2 * hi;               // lane's K pair base (even)
        v2f_t a = *(const v2f_t*)(arow + kb);     // A[m][kb], A[m][kb+1]
        v2f_t b;
        b.x = bcol[kb * NDIM];                    // B[kb][n]
        b.y = bcol[(kb + 1) * NDIM];              // B[kb+1][n]
        acc = wmma_f32(a, b, acc);
    }

    float* crow = C + (m0 + 8 * hi) * NDIM + n0 + lm;
#pragma unroll
    for (int v = 0; v < 8; ++v) crow[v * NDIM] = acc[v];
}

// P = M . Wout^T :  M (512x512), Wout (256x512) row-major -> P (512x256)
// P[i][j] = sum_k M[i][k] * Wout[j][k]  (NT gemm; Wout rows are K-contiguous)
// grid: 128 blocks * 4 waves = 512 tiles (32 x 16).
__global__ void gemm_nt_P_kernel(const float* __restrict__ M,
                                 const float* __restrict__ Wout,
                                 float* __restrict__ P) {
    const int lane = threadIdx.x & 31;
    const int wave = threadIdx.x >> 5;
    const int tile = blockIdx.x * 4 + wave;       // 0..511
    const int m0 = (tile >> 4) << 4;              // 0..496
    const int n0 = (tile & 15) << 4;              // 0..240
    const int lm = lane & 15;
    const int hi = lane >> 4;

    v8f_t acc = {};
    const float* arow = M + (m0 + lm) * NDIM;
    const float* wrow = Wout + (n0 + lm) * NDIM;  // row j of Wout = column j of B

#pragma unroll 4
    for (int kk = 0; kk < NDIM; kk += 4) {
        const int kb = kk + 2 * hi;
        v2f_t a = *(const v2f_t*)(arow + kb);
        v2f_t b = *(const v2f_t*)(wrow + kb);     // Wout[n][kb], Wout[n][kb+1]
        acc = wmma_f32(a, b, acc);
    }

    float* prow = P + (m0 + 8 * hi) * ODIM + n0 + lm;
#pragma unroll
    for (int v = 0; v < 8; ++v) prow[v * ODIM] = acc[v];
}

// Y = X . P :  X (16384x512), P (512x256) -> Y (16384x256)
// Each wave computes a 16x64 strip (4 tiles), reusing its A fragment across
// 4 WMMAs. Block = 4 waves covers a full 16x256 row strip. grid = 1024 blocks.
__global__ void gemm_out_kernel(const float* __restrict__ X,
                                const float* __restrict__ P,
                                float* __restrict__ Y) {
    const int lane = threadIdx.x & 31;
    const int wave = threadIdx.x >> 5;            // 0..3 -> n-strip
    const int m0 = blockIdx.x << 4;               // row block
    const int n0 = wave << 6;                     // 0,64,128,192
    const int lm = lane & 15;
    const int hi = lane >> 4;

    v8f_t acc0 = {}, acc1 = {}, acc2 = {}, acc3 = {};
    const float* xrow = X + (m0 + lm) * NDIM;
    const float* pcol = P + n0 + lm;

#pragma unroll 2
    for (int kk = 0; kk < NDIM; kk += 4) {
        const int kb = kk + 2 * hi;
        v2f_t a = *(const v2f_t*)(xrow + kb);     // shared A fragment
        const float* p0 = pcol + kb * ODIM;
        const float* p1 = pcol + (kb + 1) * ODIM;
        v2f_t b0 = { p0[0],  p1[0]  };
        v2f_t b1 = { p0[16], p1[16] };
        v2f_t b2 = { p0[32], p1[32] };
        v2f_t b3 = { p0[48], p1[48] };
        acc0 = wmma_f32(a, b0, acc0);
        acc1 = wmma_f32(a, b1, acc1);
        acc2 = wmma_f32(a, b2, acc2);
        acc3 = wmma_f32(a, b3, acc3);
    }

    float* yrow = Y + (m0 + 8 * hi) * ODIM + n0 + lm;
#pragma unroll
    for (int v = 0; v < 8; ++v) {
        yrow[v * ODIM + 0]  = acc0[v];
        yrow[v * ODIM + 16] = acc1[v];
        yrow[v * ODIM + 32] = acc2[v];
        yrow[v * ODIM + 48] = acc3[v];
    }
}

extern "C" void kernel_launch(void* const* d_in, const int* in_sizes, int n_in,
                              void* d_out, int out_size, void* d_ws, size_t ws_size,
                              hipStream_t stream) {
    (void)in_sizes; (void)n_in; (void)out_size; (void)ws_size;

    const float* x    = (const float*)d_in[0];   // (16384, 512)
    const float* W    = (const float*)d_in[1];   // (512, 512)
    const float* wout = (const float*)d_in[2];   // (256, 512)
    float*       out  = (float*)d_out;           // (16384, 256)

    // Workspace layout (floats): A | T0 | T1 | P
    float* A  = (float*)d_ws;
    float* T0 = A  + NDIM * NDIM;
    float* T1 = T0 + NDIM * NDIM;
    float* P  = T1 + NDIM * NDIM;                // 512*256

    // A = 0.9 I + 0.1 W
    build_A_kernel<<<(NDIM * NDIM) / 256, 256, 0, stream>>>(W, A);

    // Repeated squaring: A^2, A^4, A^8, A^16, A^32, A^64, then A^65 = A^64 * A
    gemm_nn_512_kernel<<<256, 128, 0, stream>>>(A,  A,  T0);  // A^2
    gemm_nn_512_kernel<<<256, 128, 0, stream>>>(T0, T0, T1);  // A^4
    gemm_nn_512_kernel<<<256, 128, 0, stream>>>(T1, T1, T0);  // A^8
    gemm_nn_512_kernel<<<256, 128, 0, stream>>>(T0, T0, T1);  // A^16
    gemm_nn_512_kernel<<<256, 128, 0, stream>>>(T1, T1, T0);  // A^32
    gemm_nn_512_kernel<<<256, 128, 0, stream>>>(T0, T0, T1);  // A^64
    gemm_nn_512_kernel<<<256, 128, 0, stream>>>(T1, A,  T0);  // A^65

    // P = A^65 . w_out^T   (512 x 256)
    gemm_nt_P_kernel<<<128, 128, 0, stream>>>(T0, wout, P);

    // out = x . P          (16384 x 256)
    gemm_out_kernel<<<BDIM / 16, 128, 0, stream>>>(x, P, out);
}